// Model_88115549045023
// MI455X (gfx1250) — compile-verified
//
#include <hip/hip_runtime.h>
#include <math.h>

typedef __attribute__((ext_vector_type(2))) float v2f;
typedef __attribute__((ext_vector_type(8))) float v8f;

#define B_SZ 128
#define HID  512
#define EMB  1024
#define NIN  1536   // 3*HID
#define H2   256

// ---------------- kernel 0: u_idx[b] = lower_bound(graph_batch, b) ----------------
__global__ void k_uidx(const int* __restrict__ gb, int n, int* __restrict__ uidx)
{
    int b = threadIdx.x;            // 0..127
    int lo = 0, hi = n;
    while (lo < hi) { int mid = (lo + hi) >> 1; if (gb[mid] < b) lo = mid + 1; else hi = mid; }
    uidx[b] = lo;
}

// ---------------- kernel 1: T = gather(E) @ t_w^T + t_b, routed into inp ----------
// Logical E is 384x1024:
//   rows   0..127 : desc_emb[sample[b][0]]   -> inp[b,    0:512] (+ u_feat)
//   rows 128..255 : tweet_emb[sample[b][1]]  -> inp[b,  512:1024]
//   rows 256..383 : desc_emb[sample[b][2]]   -> inp[b, 1024:1536] (+ v_feat)
// One wave per 16x16 output tile, wmma_f32_16x16x4_f32 along K=1024.
__global__ __launch_bounds__(128) void k_gemm1(
    const int* __restrict__ sample, const float* __restrict__ p_feature,
    const float* __restrict__ tweet_emb, const float* __restrict__ desc_emb,
    const float* __restrict__ t_w, const float* __restrict__ t_b,
    const int* __restrict__ uidx, float* __restrict__ inp)
{
    const int lane = threadIdx.x & 31;
    const int wave = threadIdx.x >> 5;
    const int tile = blockIdx.x * 4 + wave;     // 0..767
    const int mt = tile >> 5;                   // 0..23  (M tiles of 16 over 384 rows)
    const int nt = tile & 31;                   // 0..31  (N tiles of 16 over 512 cols)

    const int rsub = lane & 15;
    const int kb   = (lane >> 4) << 1;          // K base within 4-step: 0 or 2

    // A-fragment row gather (per-lane row pointer, fixed for the whole K loop)
    const int rowA = mt * 16 + rsub;            // 0..383
    const float* arow;
    if (rowA < 128)      arow = desc_emb  + (size_t)sample[rowA * 3 + 0]        * EMB;
    else if (rowA < 256) arow = tweet_emb + (size_t)sample[(rowA - 128) * 3 + 1] * EMB;
    else                 arow = desc_emb  + (size_t)sample[(rowA - 256) * 3 + 2] * EMB;

    // B[k,n] = t_w[n,k]  (t_w is [512,1024] row-major) -> contiguous float2 per lane
    const int colB = nt * 16 + rsub;            // 0..511
    const float* brow = t_w + (size_t)colB * EMB;

    v8f acc = {};
#pragma unroll 8
    for (int k0 = 0; k0 < EMB; k0 += 4) {
        v2f a = *(const v2f*)(arow + k0 + kb);
        v2f b = *(const v2f*)(brow + k0 + kb);
        acc = __builtin_amdgcn_wmma_f32_16x16x4_f32(false, a, false, b,
                                                    (short)0, acc, false, false);
    }

    // D layout: VGPR r -> tile row (lane>>4)*8 + r, tile col lane&15
    const int n     = nt * 16 + rsub;
    const float bias = t_b[n];
    const int mbase = mt * 16 + ((lane >> 4) << 3);

#pragma unroll
    for (int r = 0; r < 8; ++r) {
        const int m = mbase + r;                // 0..383
        float val = acc[r] + bias;
        if (m < 128) {
            val += p_feature[(size_t)uidx[m] * HID + n];
            inp[(size_t)m * NIN + n] = val;
        } else if (m < 256) {
            inp[(size_t)(m - 128) * NIN + HID + n] = val;
        } else {
            val += p_feature[(size_t)(uidx[m - 256] + 1) * HID + n];
            inp[(size_t)(m - 256) * NIN + 2 * HID + n] = val;
        }
    }
}

// ---------------- kernel 2: h = inp @ w1^T + b1  ([128x1536]x[1536x256]) ----------
__global__ __launch_bounds__(128) void k_gemm2(
    const float* __restrict__ inp, const float* __restrict__ w1,
    const float* __restrict__ b1, float* __restrict__ h)
{
    const int lane = threadIdx.x & 31;
    const int wave = threadIdx.x >> 5;
    const int tile = blockIdx.x * 4 + wave;     // 0..127
    const int mt = tile >> 4;                   // 0..7
    const int nt = tile & 15;                   // 0..15

    const int rsub = lane & 15;
    const int kb   = (lane >> 4) << 1;

    const float* arow = inp + (size_t)(mt * 16 + rsub) * NIN;
    const float* brow = w1  + (size_t)(nt * 16 + rsub) * NIN;   // B[k,n] = w1[n,k]

    v8f acc = {};
#pragma unroll 8
    for (int k0 = 0; k0 < NIN; k0 += 4) {
        v2f a = *(const v2f*)(arow + k0 + kb);
        v2f b = *(const v2f*)(brow + k0 + kb);
        acc = __builtin_amdgcn_wmma_f32_16x16x4_f32(false, a, false, b,
                                                    (short)0, acc, false, false);
    }

    const int n     = nt * 16 + rsub;
    const float bias = b1[n];
    const int mbase = mt * 16 + ((lane >> 4) << 3);
#pragma unroll
    for (int r = 0; r < 8; ++r) {
        h[(size_t)(mbase + r) * H2 + n] = acc[r] + bias;
    }
}

// ---------------- kernel 3: BatchNorm(train stats) + ReLU + GEMV + sigmoid --------
__global__ __launch_bounds__(256) void k_bnfinal(
    const float* __restrict__ h, const float* __restrict__ gamma,
    const float* __restrict__ beta, const float* __restrict__ w2,
    const float* __restrict__ b2, float* __restrict__ out)
{
    __shared__ float s_scale[H2];
    __shared__ float s_shift[H2];
    __shared__ float s_w2[H2];

    const int t = threadIdx.x;      // 0..255, one column each
    // two-pass biased batch statistics (matches mean((h-mu)^2))
    float sum = 0.f;
    for (int b = 0; b < B_SZ; ++b) sum += h[(size_t)b * H2 + t];
    const float mu = sum * (1.0f / B_SZ);
    float var = 0.f;
    for (int b = 0; b < B_SZ; ++b) {
        float d = h[(size_t)b * H2 + t] - mu;
        var += d * d;
    }
    var *= (1.0f / B_SZ);
    const float sc = gamma[t] * rsqrtf(var + 1e-5f);
    s_scale[t] = sc;
    s_shift[t] = beta[t] - mu * sc;
    s_w2[t]    = w2[t];
    __syncthreads();

    if (t < B_SZ) {
        float acc = b2[0];
        for (int n = 0; n < H2; ++n) {
            float v = h[(size_t)t * H2 + n] * s_scale[n] + s_shift[n];
            v = fmaxf(v, 0.f);
            acc += v * s_w2[n];
        }
        out[t] = 1.0f / (1.0f + expf(-acc));
    }
}

// ---------------- launcher --------------------------------------------------------
extern "C" void kernel_launch(void* const* d_in, const int* in_sizes, int n_in,
                              void* d_out, int out_size, void* d_ws, size_t ws_size,
                              hipStream_t stream)
{
    const int*   sample      = (const int*)  d_in[0];
    const int*   graph_batch = (const int*)  d_in[1];
    const float* p_feature   = (const float*)d_in[2];
    const float* tweet_emb   = (const float*)d_in[3];
    const float* desc_emb    = (const float*)d_in[4];
    const float* t_w         = (const float*)d_in[5];
    const float* t_b         = (const float*)d_in[6];
    const float* w1          = (const float*)d_in[7];
    const float* b1          = (const float*)d_in[8];
    const float* gamma       = (const float*)d_in[9];
    const float* beta        = (const float*)d_in[10];
    const float* w2          = (const float*)d_in[11];
    const float* b2          = (const float*)d_in[12];
    float* out = (float*)d_out;

    char* ws = (char*)d_ws;
    int*   uidx = (int*)ws;                                    // 128 ints
    float* inp  = (float*)(ws + 512);                          // 128*1536 f32
    float* h    = (float*)(ws + 512 + (size_t)B_SZ * NIN * 4); // 128*256 f32

    const int n_nodes = in_sizes[1];

    k_uidx  <<<1,   128, 0, stream>>>(graph_batch, n_nodes, uidx);
    k_gemm1 <<<192, 128, 0, stream>>>(sample, p_feature, tweet_emb, desc_emb,
                                      t_w, t_b, uidx, inp);
    k_gemm2 <<<32,  128, 0, stream>>>(inp, w1, b1, h);
    k_bnfinal<<<1,  256, 0, stream>>>(h, gamma, beta, w2, b2, out);
}